// GraphDistanceLoss_sim_matrix_69097433858449
// MI455X (gfx1250) — compile-verified
//
#include <hip/hip_runtime.h>

typedef __attribute__((ext_vector_type(2))) float v2f;
typedef __attribute__((ext_vector_type(8))) float v8f;

#define U_ROWS     8192
#define DIM        512
#define NPAIR      131072
#define NTILES     (NPAIR / 16)      // 8192 tiles, one wave each
#define AMP_F      4.0f
#define REG_LAMBDA 0.1f
#define EPS_F      1e-12f

// ---------------------------------------------------------------------------
// K0: per-row L2 norm (clamped) + fused regularization partial:
//     norms[r]      = max(||emb[r]||, EPS)
//     regPartial[r] = sum_d (emb[r][d]/norms[r] - orig[r][d])^2
// one 256-thread block per row (each thread handles 2 of the 512 floats)
// ---------------------------------------------------------------------------
__global__ __launch_bounds__(256) void norm_reg_kernel(
    const float* __restrict__ emb, const float* __restrict__ orig,
    float* __restrict__ norms, float* __restrict__ regPartial) {
  __shared__ float red[8];
  __shared__ float bnorm;
  const int row  = blockIdx.x;
  const int tid  = threadIdx.x;
  const int wave = tid >> 5;
  const int lane = tid & 31;

  const float2 e = ((const float2*)(emb + (size_t)row * DIM))[tid];
  float ss = e.x * e.x + e.y * e.y;
  #pragma unroll
  for (int off = 16; off > 0; off >>= 1) ss += __shfl_xor(ss, off, 32);
  if (lane == 0) red[wave] = ss;
  __syncthreads();
  if (tid == 0) {
    float t = 0.f;
    #pragma unroll
    for (int w = 0; w < 8; ++w) t += red[w];
    float n = fmaxf(sqrtf(t), EPS_F);
    norms[row] = n;
    bnorm = n;
  }
  __syncthreads();
  const float n = bnorm;
  const float2 o = ((const float2*)(orig + (size_t)row * DIM))[tid];
  const float dx = e.x / n - o.x;
  const float dy = e.y / n - o.y;
  float rs = dx * dx + dy * dy;
  #pragma unroll
  for (int off = 16; off > 0; off >>= 1) rs += __shfl_xor(rs, off, 32);
  __syncthreads();   // red[] reuse
  if (lane == 0) red[wave] = rs;
  __syncthreads();
  if (tid == 0) {
    float t = 0.f;
    #pragma unroll
    for (int w = 0; w < 8; ++w) t += red[w];
    regPartial[row] = t;
  }
}

// ---------------------------------------------------------------------------
// K1: one wave32 per 16-pair tile. D = A(16xK) * B(Kx16) via 128 accumulating
// v_wmma_f32_16x16x4_f32 steps; diagonal of D = the 16 pair dot products.
// A layout (32-bit 16x4): lanes 0-15 carry (K=0,K=1), lanes 16-31 (K=2,K=3),
// row M = lane%16. B (4x16) is symmetric with N = lane%16. So each lane loads
// one float2 from emb[row] at offset k0 + 2*(lane/16) for each matrix.
// cur_sim = diag / (n_i * n_j)  (== dot of normalized rows).
// ---------------------------------------------------------------------------
__global__ __launch_bounds__(256) void pair_wmma_kernel(
    const float* __restrict__ emb, const float* __restrict__ sim,
    const int* __restrict__ pair_idx, const float* __restrict__ norms,
    float* __restrict__ pairPartial) {
  const int tid  = threadIdx.x;
  const int lane = tid & 31;
  const int tile = blockIdx.x * 8 + (tid >> 5);   // 8 waves per block
  const int p0   = tile * 16;
  const int l16  = lane & 15;
  const int half = lane >> 4;

  const int ri = pair_idx[(size_t)(p0 + l16) * 2 + 0];
  const int rj = pair_idx[(size_t)(p0 + l16) * 2 + 1];

  const float2* pa = (const float2*)(emb + (size_t)ri * DIM + 2 * half);
  const float2* pb = (const float2*)(emb + (size_t)rj * DIM + 2 * half);

  v8f c = {0.f, 0.f, 0.f, 0.f, 0.f, 0.f, 0.f, 0.f};
  #pragma unroll 4
  for (int k = 0; k < DIM / 4; ++k) {        // 128 WMMA steps, K advances by 4
    const float2 af = pa[2 * k];
    const float2 bf = pb[2 * k];
    v2f a; a[0] = af.x; a[1] = af.y;
    v2f b; b[0] = bf.x; b[1] = bf.y;
    c = __builtin_amdgcn_wmma_f32_16x16x4_f32(
        /*neg_a=*/false, a, /*neg_b=*/false, b,
        /*c_mod=*/(short)0, c, /*reuse_a=*/false, /*reuse_b=*/false);
  }

  // Diagonal: pair p<8 -> (vgpr p, lane p); pair p>=8 -> (vgpr p-8, lane p+16)
  float diag = 0.f;
  #pragma unroll
  for (int v = 0; v < 8; ++v) {
    const float cv = c[v];
    if (lane == v || lane == v + 24) diag = cv;
  }
  const bool writer = (lane < 8) || (lane >= 24);
  float contrib = 0.f;
  if (writer) {
    const float ni  = norms[ri];
    const float nj  = norms[rj];
    const float cur = diag / (ni * nj);
    const float tgt = sim[(size_t)ri * U_ROWS + rj];
    const float d   = fabsf(cur - tgt) * AMP_F;
    contrib = d * d;
  }
  #pragma unroll
  for (int off = 16; off > 0; off >>= 1) contrib += __shfl_xor(contrib, off, 32);
  if (lane == 0) pairPartial[tile] = contrib;
}

// ---------------------------------------------------------------------------
// K2: deterministic single-block reduction of 8192 pair partials and 8192 reg
// partials -> (total, loss, reg)
// ---------------------------------------------------------------------------
__global__ __launch_bounds__(256) void finalize_kernel(
    const float* __restrict__ regPartial, const float* __restrict__ pairPartial,
    float* __restrict__ out) {
  __shared__ float sp[256];
  __shared__ float sr[256];
  const int t = threadIdx.x;
  float ap = 0.f, ar = 0.f;
  for (int k = t; k < NTILES; k += 256) ap += pairPartial[k];
  for (int k = t; k < U_ROWS; k += 256) ar += regPartial[k];
  sp[t] = ap; sr[t] = ar;
  __syncthreads();
  #pragma unroll
  for (int s = 128; s > 0; s >>= 1) {
    if (t < s) { sp[t] += sp[t + s]; sr[t] += sr[t + s]; }
    __syncthreads();
  }
  if (t == 0) {
    const float loss = sp[0] / (float)NPAIR;
    const float reg  = REG_LAMBDA * (sr[0] / (float)U_ROWS);
    out[0] = loss + reg;
    out[1] = loss;
    out[2] = reg;
  }
}

extern "C" void kernel_launch(void* const* d_in, const int* in_sizes, int n_in,
                              void* d_out, int out_size, void* d_ws, size_t ws_size,
                              hipStream_t stream) {
  (void)in_sizes; (void)n_in; (void)out_size; (void)ws_size;
  const float* emb  = (const float*)d_in[0];
  const float* orig = (const float*)d_in[1];
  const float* sim  = (const float*)d_in[2];
  const int*   pidx = (const int*)d_in[3];

  float* ws     = (float*)d_ws;        // ~96 KB of scratch used
  float* norms  = ws;                  // U_ROWS floats
  float* regP   = ws + U_ROWS;         // U_ROWS floats
  float* pairP  = ws + 2 * U_ROWS;     // NTILES floats

  norm_reg_kernel<<<U_ROWS, 256, 0, stream>>>(emb, orig, norms, regP);
  pair_wmma_kernel<<<NTILES / 8, 256, 0, stream>>>(emb, sim, pidx, norms, pairP);
  finalize_kernel<<<1, 256, 0, stream>>>(regP, pairP, (float*)d_out);
}